// GCN_19825569038523
// MI455X (gfx1250) — compile-verified
//
#include <hip/hip_runtime.h>

#define N_NODES 100000
#define N_EDGES 1600000
#define DIN 128
#define DH 64
#define LN_EPS 1e-5f

typedef __attribute__((ext_vector_type(2))) float v2f;
typedef __attribute__((ext_vector_type(8))) float v8f;

// ---------------------------------------------------------------------------
// Degree accumulation: deg[idx[i]] += 1  (atomic, L2-resident target)
// ---------------------------------------------------------------------------
__global__ __launch_bounds__(256) void degree_kernel(const int* __restrict__ idx,
                                                     float* __restrict__ deg, int e) {
    int i = blockIdx.x * blockDim.x + threadIdx.x;
    if (i < e) atomicAdd(&deg[idx[i]], 1.0f);
}

// ns = rsqrt(max(deg_out,1)), nd = rsqrt(max(deg_in,1))  (in place)
__global__ __launch_bounds__(256) void finalize_norm_kernel(float* __restrict__ ns,
                                                            float* __restrict__ nd, int n) {
    int i = blockIdx.x * blockDim.x + threadIdx.x;
    if (i < n) {
        ns[i] = rsqrtf(fmaxf(ns[i], 1.0f));
        nd[i] = rsqrtf(fmaxf(nd[i], 1.0f));
    }
}

// ---------------------------------------------------------------------------
// H = (X * ns[row]) @ W   via V_WMMA_F32_16X16X4_F32 (fp32 exact path).
// Block: 128 threads = 4 waves. Block owns a 16-row stripe; wave w owns the
// 16-column tile [w*16, w*16+16). K-loop steps of 4.
// A frag (16x4): lanes 0-15 -> rows, VGPR0/1 = K {0,1} (lanes<16) / {2,3} (lanes>=16)
// B frag (4x16): lanes 0-15 -> cols, VGPR0/1 = K {0,1} / {2,3}
// C/D (16x16):   lane<16: M=vgpr, N=lane ; lane>=16: M=8+vgpr, N=lane-16
// ---------------------------------------------------------------------------
template <int K>
__global__ __launch_bounds__(128) void gemm_ns_wmma(const float* __restrict__ X,
                                                    const float* __restrict__ W,
                                                    const float* __restrict__ ns,
                                                    float* __restrict__ H) {
    const int lane    = threadIdx.x & 31;
    const int wave    = threadIdx.x >> 5;
    const int rowBase = blockIdx.x * 16;
    const int colBase = wave * 16;
    const int half    = lane >> 4;       // 0: K+{0,1}, 1: K+{2,3}
    const int l15     = lane & 15;
    const int row     = rowBase + l15;   // A-matrix row for this lane
    const int col     = colBase + l15;   // B/C-matrix column for this lane
    const float s     = ns[row];         // fused row scaling (feats * norm_src)

    v8f c = {};
#pragma unroll 4
    for (int k0 = 0; k0 < K; k0 += 4) {
        const int kk = k0 + half * 2;
        const float2 xa = *(const float2*)(X + row * K + kk);
        v2f a;
        a.x = xa.x * s;
        a.y = xa.y * s;
        v2f b;
        b.x = W[kk * DH + col];
        b.y = W[(kk + 1) * DH + col];
        // D = A*B + C  (8 args: neg_a, A, neg_b, B, c_mod, C, reuse_a, reuse_b)
        c = __builtin_amdgcn_wmma_f32_16x16x4_f32(false, a, false, b, (short)0, c,
                                                  false, false);
    }

    const int mBase = half * 8;
#pragma unroll
    for (int i = 0; i < 8; ++i) {
        H[(size_t)(rowBase + mBase + i) * DH + col] = c[i];
    }
}

// ---------------------------------------------------------------------------
// Edge aggregation: agg[dst[e], :] += H[src[e], :]
// One wave32 per edge; lane moves float2 (2 of 64 cols). Index loads are
// wave-uniform -> readfirstlane pins the row bases in SGPRs. H and agg are
// 25.6 MB each: resident in the 192 MB L2, so gathers + f32 atomics stay in L2.
// Grid divides E exactly (no bounds check, no divergence).
// ---------------------------------------------------------------------------
__global__ __launch_bounds__(256) void aggregate_kernel(const float* __restrict__ H,
                                                        const int* __restrict__ src,
                                                        const int* __restrict__ dst,
                                                        float* __restrict__ agg) {
    const int gid  = blockIdx.x * blockDim.x + threadIdx.x;
    const int e    = gid >> 5;
    const int lane = gid & 31;
    const int s = __builtin_amdgcn_readfirstlane(src[e]);
    const int d = __builtin_amdgcn_readfirstlane(dst[e]);
    const float2 v = *(const float2*)(H + (size_t)s * DH + lane * 2);
    float* p = agg + (size_t)d * DH + lane * 2;
    atomicAdd(p + 0, v.x);
    atomicAdd(p + 1, v.y);
}

// ---------------------------------------------------------------------------
// Post: v = agg*nd + b  [-> ReLU]  [-> write pre-norm copy]  [-> LayerNorm]
// One wave per node (2 elements/lane over DH=64); shfl_xor tree reduction.
// ---------------------------------------------------------------------------
__global__ __launch_bounds__(256) void post_kernel(const float* __restrict__ agg,
                                                   const float* __restrict__ nd,
                                                   const float* __restrict__ bias,
                                                   const float* __restrict__ g,
                                                   const float* __restrict__ be,
                                                   float* __restrict__ outx,
                                                   float* __restrict__ pre,
                                                   int n, int do_relu, int do_ln) {
    const int node = blockIdx.x * 8 + (threadIdx.x >> 5);
    if (node >= n) return;
    const int lane = threadIdx.x & 31;
    const int c0   = lane * 2;
    const float ndv = nd[node];
    const float* r = agg + (size_t)node * DH;
    float v0 = r[c0 + 0] * ndv + bias[c0 + 0];
    float v1 = r[c0 + 1] * ndv + bias[c0 + 1];
    if (do_relu) {
        v0 = fmaxf(v0, 0.0f);
        v1 = fmaxf(v1, 0.0f);
    }
    if (pre) {
        pre[(size_t)node * DH + c0 + 0] = v0;
        pre[(size_t)node * DH + c0 + 1] = v1;
    }
    if (do_ln) {
        float ssum = v0 + v1;
#pragma unroll
        for (int off = 16; off >= 1; off >>= 1) ssum += __shfl_xor(ssum, off, 32);
        const float mu = ssum * (1.0f / DH);
        const float d0 = v0 - mu, d1 = v1 - mu;
        float vs = d0 * d0 + d1 * d1;
#pragma unroll
        for (int off = 16; off >= 1; off >>= 1) vs += __shfl_xor(vs, off, 32);
        const float inv = rsqrtf(vs * (1.0f / DH) + LN_EPS);
        v0 = d0 * inv * g[c0 + 0] + be[c0 + 0];
        v1 = d1 * inv * g[c0 + 1] + be[c0 + 1];
    }
    float* o = outx + (size_t)node * DH + c0;
    o[0] = v0;
    o[1] = v1;
}

// ---------------------------------------------------------------------------
extern "C" void kernel_launch(void* const* d_in, const int* in_sizes, int n_in,
                              void* d_out, int out_size, void* d_ws, size_t ws_size,
                              hipStream_t stream) {
    (void)in_sizes; (void)n_in; (void)out_size; (void)ws_size;

    const float* feats = (const float*)d_in[0];
    const float* W1    = (const float*)d_in[1];
    const float* b1    = (const float*)d_in[2];
    const float* g1    = (const float*)d_in[3];
    const float* be1   = (const float*)d_in[4];
    const float* W2    = (const float*)d_in[5];
    const float* b2    = (const float*)d_in[6];
    const float* g2    = (const float*)d_in[7];
    const float* be2   = (const float*)d_in[8];
    const float* W3    = (const float*)d_in[9];
    const float* b3    = (const float*)d_in[10];
    const int*   src   = (const int*)d_in[11];
    const int*   dst   = (const int*)d_in[12];

    // workspace layout (fp32): ns[N] | nd[N] | H[N*64] | AGG[N*64] | XB[N*64]
    float* ns  = (float*)d_ws;
    float* nd  = ns + N_NODES;
    float* H   = nd + N_NODES;
    float* AGG = H + (size_t)N_NODES * DH;
    float* XB  = AGG + (size_t)N_NODES * DH;

    float* h_emb = (float*)d_out;                       // [N, 64] pre-norm layer-2
    float* h_out = h_emb + (size_t)N_NODES * DH;        // [N, 64] final layer

    // degrees -> rsqrt norms
    hipMemsetAsync(ns, 0, 2 * (size_t)N_NODES * sizeof(float), stream);
    degree_kernel<<<N_EDGES / 256, 256, 0, stream>>>(src, ns, N_EDGES);
    degree_kernel<<<N_EDGES / 256, 256, 0, stream>>>(dst, nd, N_EDGES);
    finalize_norm_kernel<<<(N_NODES + 255) / 256, 256, 0, stream>>>(ns, nd, N_NODES);

    const size_t aggBytes = (size_t)N_NODES * DH * sizeof(float);

    // ---- layer 1: (feats*ns)@W1 -> agg -> *nd+b1 -> ReLU -> LN ----
    gemm_ns_wmma<DIN><<<N_NODES / 16, 128, 0, stream>>>(feats, W1, ns, H);
    hipMemsetAsync(AGG, 0, aggBytes, stream);
    aggregate_kernel<<<N_EDGES / 8, 256, 0, stream>>>(H, src, dst, AGG);
    post_kernel<<<N_NODES / 8, 256, 0, stream>>>(AGG, nd, b1, g1, be1, XB, nullptr,
                                                 N_NODES, 1, 1);

    // ---- layer 2: emits h_emb (pre-norm) and normalized XB ----
    gemm_ns_wmma<DH><<<N_NODES / 16, 128, 0, stream>>>(XB, W2, ns, H);
    hipMemsetAsync(AGG, 0, aggBytes, stream);
    aggregate_kernel<<<N_EDGES / 8, 256, 0, stream>>>(H, src, dst, AGG);
    post_kernel<<<N_NODES / 8, 256, 0, stream>>>(AGG, nd, b2, g2, be2, XB, h_emb,
                                                 N_NODES, 1, 1);

    // ---- layer 3: no activation, no norm ----
    gemm_ns_wmma<DH><<<N_NODES / 16, 128, 0, stream>>>(XB, W3, ns, H);
    hipMemsetAsync(AGG, 0, aggBytes, stream);
    aggregate_kernel<<<N_EDGES / 8, 256, 0, stream>>>(H, src, dst, AGG);
    post_kernel<<<N_NODES / 8, 256, 0, stream>>>(AGG, nd, b3, nullptr, nullptr, h_out,
                                                 nullptr, N_NODES, 0, 0);
}